// GLMAttention_87479893885573
// MI455X (gfx1250) — compile-verified
//
#include <hip/hip_runtime.h>
#include <hip/hip_bf16.h>

// ---------------------------------------------------------------------------
// GLM GQA paged-attention decode for MI455X (gfx1250, wave32, WMMA).
// Memory-bound (~430MB / ~7.4GFLOP, AI ~17 FLOP/B -> ~19us HBM floor):
//  * fp32 end-to-end via V_WMMA_F32_16X16X4_F32 (matches reference numerics)
//  * flash-decode with 8-way sequence partitioning: 64*2*8 = 1024 waves
//  * paged K/V streamed with GLOBAL_LOAD_ASYNC_TO_LDS_B128, double-buffered
//    per 16-token cache block, synchronized with s_wait_asynccnt
// ---------------------------------------------------------------------------

#define H    4096
#define NH   32
#define HD   128
#define NKV  2
#define G    16          // query heads per KV group
#define BS   16          // paged block size == WMMA tile N  (!)
#define MB   256
#define BATCH 64
#define SPLIT 8
#define BLKS_PER_PART (MB / SPLIT)      // 32 cache blocks = 512 tokens
#define KVW  (2 * NKV * HD)             // 512
#define SCALE 0.08838834764831845f      // 1/sqrt(128)

#define LPITCH 132                      // LDS row pitch (floats): pad 128->132
#define LPITCHB (LPITCH * 4)            // 528 bytes

typedef __attribute__((ext_vector_type(2))) float v2f;
typedef __attribute__((ext_vector_type(8))) float v8f;

// D = A(16x4 f32) * B(4x16 f32) + C(16x16 f32), one tile per wave32.
__device__ __forceinline__ v8f wmma_f32(v2f a, v2f b, v8f c) {
    return __builtin_amdgcn_wmma_f32_16x16x4_f32(
        /*neg_a=*/false, a, /*neg_b=*/false, b,
        /*c_mod=*/(short)0, c, /*reuse_a=*/false, /*reuse_b=*/false);
}

// Async-copy one 16-token K block + V block (512B per token row) into LDS.
// Each instruction: 32 lanes x 16B = 512B = one token's head_dim row.
// Tracked with ASYNCcnt (32 ops per call).
__device__ __forceinline__ void issue_kv_async(const float* Kp, const float* Vp,
                                               const float* kls, const float* vls,
                                               int lane)
{
    const uint32_t kbase = (uint32_t)(uintptr_t)kls + lane * 16;
    const uint32_t vbase = (uint32_t)(uintptr_t)vls + lane * 16;
    const uint64_t gk = (uint64_t)(uintptr_t)Kp + lane * 16;
    const uint64_t gv = (uint64_t)(uintptr_t)Vp + lane * 16;
    #pragma unroll
    for (int t = 0; t < BS; ++t) {
        const uint64_t go = (uint64_t)(t * NKV * HD * 4);   // token stride in cache
        asm volatile("global_load_async_to_lds_b128 %0, %1, off"
                     :: "v"(kbase + t * LPITCHB), "v"(gk + go) : "memory");
        asm volatile("global_load_async_to_lds_b128 %0, %1, off"
                     :: "v"(vbase + t * LPITCHB), "v"(gv + go) : "memory");
    }
}

// ---------------------------------------------------------------------------
// Generic C[64,N] = A[64,4096] @ W[4096,N] + bias[N]  (fp32 WMMA).
// 8 waves / block, each wave owns one 16x16 output tile; K-loop step 4.
// ---------------------------------------------------------------------------
__global__ __launch_bounds__(256) void gemm_bias_kernel(
    const float* __restrict__ A, const float* __restrict__ W,
    const float* __restrict__ bias, float* __restrict__ C, int N)
{
    const int lane = threadIdx.x & 31;
    const int wave = threadIdx.x >> 5;
    const int hi   = lane >> 4;       // half-wave select
    const int lm   = lane & 15;
    const int m0   = blockIdx.y * 16;
    const int n0   = (blockIdx.x * 8 + wave) * 16;
    if (n0 >= N) return;              // wave-uniform

    const float* arow = A + (size_t)(m0 + lm) * H;   // A: lane holds row M=lm
    const float* wcol = W + n0 + lm;                 // B: lane holds col N=lm

    v8f acc = {};
    #pragma unroll 8
    for (int k0 = 0; k0 < H; k0 += 4) {
        const int ka = k0 + 2 * hi;
        v2f a; a.x = arow[ka];                  a.y = arow[ka + 1];
        v2f b; b.x = wcol[(size_t)ka * N];      b.y = wcol[(size_t)(ka + 1) * N];
        acc = wmma_f32(a, b, acc);
    }

    const float bi = bias[n0 + lm];
    #pragma unroll
    for (int r = 0; r < 8; ++r) {               // C row = r + 8*hi
        const int m = m0 + r + 8 * hi;
        C[(size_t)m * N + n0 + lm] = acc[r] + bi;
    }
}

// ---------------------------------------------------------------------------
// Flash-decode partition kernel. One wave per (seq b, kv head, partition).
// K/V blocks stream global->LDS via async b128 copies (double-buffered);
// scores Q[16x128] x K^T via 32 WMMAs; softmax stats via 16-lane shfl_xor;
// P transposed through padded LDS; P*V via 8x4 WMMAs into 8 v8f accumulators.
// New token K/V (pos = ctx-1) substituted from the fresh KV projection.
// ---------------------------------------------------------------------------
__global__ __launch_bounds__(32) void attn_part_kernel(
    const float* __restrict__ q_ws, const float* __restrict__ kv_ws,
    const float* __restrict__ key_cache, const float* __restrict__ value_cache,
    const int* __restrict__ block_tables, const int* __restrict__ context_lens,
    float* __restrict__ pO, float* __restrict__ pM, float* __restrict__ pL)
{
    const int b    = blockIdx.x;          // 64
    const int kvh  = blockIdx.y;          // 2
    const int part = blockIdx.z;          // 8
    const int lane = threadIdx.x;
    const int hi   = lane >> 4;
    const int lm   = lane & 15;

    const int ctx = context_lens[b];
    const int pos = ctx - 1;
    const int tok_start = part * BLKS_PER_PART * BS;

    const size_t poBase  = ((size_t)(b * NKV + kvh) * SPLIT + part) * G * HD;
    const int    pmlBase = ((b * NKV + kvh) * SPLIT + part) * G;

    if (tok_start >= ctx) {               // empty partition (wave-uniform)
        if (lm == 0) {
            #pragma unroll
            for (int r = 0; r < 8; ++r) {
                pM[pmlBase + r + 8 * hi] = -1e30f;
                pL[pmlBase + r + 8 * hi] = 0.0f;
            }
        }
        #pragma unroll
        for (int nt = 0; nt < 8; ++nt)
            #pragma unroll
            for (int r = 0; r < 8; ++r)
                pO[poBase + (size_t)(r + 8 * hi) * HD + nt * 16 + lm] = 0.0f;
        return;
    }

    // double-buffered K/V stage: [buf][K|V][16 tokens * 132-float pitch]
    __shared__ __align__(16) float kvstage[2][2][BS * LPITCH];
    __shared__ float pbuf[16 * 17];       // padded P tile (one wave / block)

    const float* qp   = q_ws + (size_t)b * H + (kvh * G + lm) * HD;  // Q row lm
    const float* knew = kv_ws + (size_t)b * KVW + kvh * HD;
    const float* vnew = kv_ws + (size_t)b * KVW + NKV * HD + kvh * HD;
    const int*   btp  = block_tables + b * MB + part * BLKS_PER_PART;

    float mrow[8], lrow[8];
    v8f   oacc[8];
    #pragma unroll
    for (int r = 0; r < 8; ++r) { mrow[r] = -1e30f; lrow[r] = 0.0f; }
    #pragma unroll
    for (int nt = 0; nt < 8; ++nt) oacc[nt] = (v8f){};

    int nblocks = (ctx - tok_start + BS - 1) / BS;
    if (nblocks > BLKS_PER_PART) nblocks = BLKS_PER_PART;

    // prime the pipeline: async-copy block 0 into buffer 0
    {
        const int blk0 = btp[0];
        issue_kv_async(key_cache   + (((size_t)blk0 * BS) * NKV + kvh) * HD,
                       value_cache + (((size_t)blk0 * BS) * NKV + kvh) * HD,
                       &kvstage[0][0][0], &kvstage[0][1][0], lane);
    }

    for (int cb = 0; cb < nblocks; ++cb) {
        const int tbase = tok_start + cb * BS;
        const bool hasNext = (cb + 1 < nblocks);
        if (hasNext) {                    // overlap next block's copy with compute
            const int nb = btp[cb + 1];
            issue_kv_async(key_cache   + (((size_t)nb * BS) * NKV + kvh) * HD,
                           value_cache + (((size_t)nb * BS) * NKV + kvh) * HD,
                           &kvstage[(cb + 1) & 1][0][0],
                           &kvstage[(cb + 1) & 1][1][0], lane);
            asm volatile("s_wait_asynccnt 0x20" ::: "memory");  // current buf done
        } else {
            asm volatile("s_wait_asynccnt 0x0" ::: "memory");
        }
        const float* kls = &kvstage[cb & 1][0][0];
        const float* vls = &kvstage[cb & 1][1][0];

        // ---- scores tile: C[16 heads x 16 tokens] ----
        const int   mytok = tbase + lm;                  // this lane's token (B col)
        const float* krow = kls + lm * LPITCH;           // token row, conflict-free
        const bool  repl  = (mytok == pos);
        v8f c = {};
        #pragma unroll
        for (int k = 0; k < 32; ++k) {
            const int d = 4 * k + 2 * hi;
            v2f qa; qa.x = qp[d];   qa.y = qp[d + 1];
            v2f kb; kb.x = krow[d]; kb.y = krow[d + 1];
            if (repl) { kb.x = knew[d]; kb.y = knew[d + 1]; }
            c = wmma_f32(qa, kb, c);
        }

        // ---- masked online softmax ----
        const float maskadd = (mytok < ctx) ? 0.0f : -1e30f;
        float f[8];
        #pragma unroll
        for (int r = 0; r < 8; ++r) c[r] = c[r] * SCALE + maskadd;
        #pragma unroll
        for (int r = 0; r < 8; ++r) {                    // row max over 16 lanes
            float v = c[r];
            v = fmaxf(v, __shfl_xor(v, 1, 16));
            v = fmaxf(v, __shfl_xor(v, 2, 16));
            v = fmaxf(v, __shfl_xor(v, 4, 16));
            v = fmaxf(v, __shfl_xor(v, 8, 16));
            const float mnew = fmaxf(mrow[r], v);
            f[r] = __expf(mrow[r] - mnew);
            mrow[r] = mnew;
            const float p = __expf(c[r] - mnew);
            c[r] = p;
            lrow[r] = lrow[r] * f[r] + p;                // per-lane partial rowsum
        }

        // ---- P: C-layout -> A-layout via padded LDS (wave-local, in-order) ----
        #pragma unroll
        for (int r = 0; r < 8; ++r) pbuf[(r + 8 * hi) * 17 + lm] = c[r];
        v2f pa[4];
        #pragma unroll
        for (int k = 0; k < 4; ++k) {
            const int t = 4 * k + 2 * hi;
            pa[k].x = pbuf[lm * 17 + t];
            pa[k].y = pbuf[lm * 17 + t + 1];
        }

        // ---- O[16 x 128] += P[16 x 16] * V_blk[16 x 128] ----
        #pragma unroll
        for (int nt = 0; nt < 8; ++nt) {
            v8f o = oacc[nt];
            #pragma unroll
            for (int r = 0; r < 8; ++r) o[r] *= f[r];    // rescale by exp(mold-mnew)
            #pragma unroll
            for (int k = 0; k < 4; ++k) {
                const int t0 = 4 * k + 2 * hi;
                v2f vb;
                vb.x = vls[t0 * LPITCH + nt * 16 + lm];
                vb.y = vls[(t0 + 1) * LPITCH + nt * 16 + lm];
                if (tbase + t0     == pos) vb.x = vnew[nt * 16 + lm];
                if (tbase + t0 + 1 == pos) vb.y = vnew[nt * 16 + lm];
                o = wmma_f32(pa[k], vb, o);
            }
            oacc[nt] = o;
        }
    }

    // ---- finalize: reduce rowsums, emit (m, l, O_unnormalized) ----
    #pragma unroll
    for (int r = 0; r < 8; ++r) {
        float v = lrow[r];
        v += __shfl_xor(v, 1, 16);
        v += __shfl_xor(v, 2, 16);
        v += __shfl_xor(v, 4, 16);
        v += __shfl_xor(v, 8, 16);
        lrow[r] = v;
    }
    if (lm == 0) {
        #pragma unroll
        for (int r = 0; r < 8; ++r) {
            pM[pmlBase + r + 8 * hi] = mrow[r];
            pL[pmlBase + r + 8 * hi] = lrow[r];
        }
    }
    #pragma unroll
    for (int nt = 0; nt < 8; ++nt)
        #pragma unroll
        for (int r = 0; r < 8; ++r)
            pO[poBase + (size_t)(r + 8 * hi) * HD + nt * 16 + lm] = oacc[nt][r];
}

// ---------------------------------------------------------------------------
// Merge the SPLIT partial (m, l, O) triplets -> normalized attention output.
// ---------------------------------------------------------------------------
__global__ __launch_bounds__(256) void combine_kernel(
    const float* __restrict__ pO, const float* __restrict__ pM,
    const float* __restrict__ pL, float* __restrict__ attn_ws)
{
    const int idx = blockIdx.x * 256 + threadIdx.x;      // b*H + h*HD + d
    const int d   = idx & (HD - 1);
    const int g   = (idx >> 7) & (G - 1);
    const int kvh = (idx >> 11) & (NKV - 1);
    const int b   = idx >> 12;
    const int base = ((b * NKV + kvh) * SPLIT) * G + g;

    float ms[SPLIT];
    float M = -1e30f;
    #pragma unroll
    for (int p = 0; p < SPLIT; ++p) { ms[p] = pM[base + p * G]; M = fmaxf(M, ms[p]); }
    float L = 0.0f, o = 0.0f;
    #pragma unroll
    for (int p = 0; p < SPLIT; ++p) {
        const float w = __expf(ms[p] - M);
        L += w * pL[base + p * G];
        o += w * pO[(size_t)(base + p * G) * HD + d];
    }
    attn_ws[idx] = o / L;
}

// ---------------------------------------------------------------------------
extern "C" void kernel_launch(void* const* d_in, const int* in_sizes, int n_in,
                              void* d_out, int out_size, void* d_ws, size_t ws_size,
                              hipStream_t stream)
{
    const float* hs  = (const float*)d_in[0];
    const float* wq  = (const float*)d_in[1];
    const float* bq  = (const float*)d_in[2];
    const float* wkv = (const float*)d_in[3];
    const float* bkv = (const float*)d_in[4];
    const float* wo  = (const float*)d_in[5];
    const float* bo  = (const float*)d_in[6];
    const float* kc  = (const float*)d_in[7];
    const float* vc  = (const float*)d_in[8];
    const int*   bt  = (const int*)d_in[9];
    const int*   cl  = (const int*)d_in[10];
    float*       out = (float*)d_out;

    // workspace carve-up (floats): ~10.75 MB total
    float* q_ws    = (float*)d_ws;                     // 64*4096
    float* kv_ws   = q_ws    + (size_t)BATCH * H;      // 64*512
    float* attn_ws = kv_ws   + (size_t)BATCH * KVW;    // 64*4096
    float* pO      = attn_ws + (size_t)BATCH * H;      // 64*2*8*16*128
    float* pM      = pO + (size_t)BATCH * NKV * SPLIT * G * HD;  // 64*2*8*16
    float* pL      = pM + (size_t)BATCH * NKV * SPLIT * G;

    // 1) Q = hs @ wq + bq        [64,4096]
    gemm_bias_kernel<<<dim3(H / 128, BATCH / 16), 256, 0, stream>>>(hs, wq, bq, q_ws, H);
    // 2) KV = hs @ wkv + bkv     [64,512]
    gemm_bias_kernel<<<dim3(KVW / 128, BATCH / 16), 256, 0, stream>>>(hs, wkv, bkv, kv_ws, KVW);
    // 3) partitioned flash-decode attention: 64*2*8 = 1024 waves
    attn_part_kernel<<<dim3(BATCH, NKV, SPLIT), 32, 0, stream>>>(
        q_ws, kv_ws, kc, vc, bt, cl, pO, pM, pL);
    // 4) merge partitions
    combine_kernel<<<dim3((BATCH * H) / 256), 256, 0, stream>>>(pO, pM, pL, attn_ws);
    // 5) out = attn @ wo + bo    [64,4096]
    gemm_bias_kernel<<<dim3(H / 128, BATCH / 16), 256, 0, stream>>>(attn_ws, wo, bo, out, H);
}